// CrownSAGE_29489245454480
// MI455X (gfx1250) — compile-verified
//
#include <hip/hip_runtime.h>

// ---------------------------------------------------------------------------
// CrownSAGE on gfx1250 (MI455X): 2-layer GraphSAGE
//   layer: out = relu?( mean_agg(x)@W_l + x@W_r + b )
// Dense part uses V_WMMA_F32_16X16X4_F32 (exact fp32 matrix pipe).
// Sparse part uses GLOBAL_ATOMIC_ADD_F32 scatter (L2-resident agg buffers).
// ---------------------------------------------------------------------------

typedef float v2f __attribute__((ext_vector_type(2)));
typedef float v8f __attribute__((ext_vector_type(8)));

// ---- degree count: deg[dst[e]] += 1 ---------------------------------------
__global__ void sage_deg_kernel(const int* __restrict__ dst,
                                float* __restrict__ deg, int E) {
  int e = blockIdx.x * blockDim.x + threadIdx.x;
  if (e < E) atomicAdd(&deg[dst[e]], 1.0f);
}

// ---- scatter-add features: agg[dst[e]][:] += feat[src[e]][:] --------------
// One thread per (edge, 4-float chunk): float4 gather + 4 f32 atomics.
template <int F>
__global__ void sage_scatter_kernel(const float* __restrict__ feat,
                                    const int* __restrict__ src,
                                    const int* __restrict__ dst,
                                    float* __restrict__ agg, int E) {
  constexpr int CH = F / 4;
  int t = blockIdx.x * blockDim.x + threadIdx.x;
  if (t >= E * CH) return;
  int e = t / CH;
  int c = (t % CH) * 4;
  int s = src[e];
  int d = dst[e];
  const float4 v = *(const float4*)(feat + (long long)s * F + c);
  float* p = agg + (long long)d * F + c;
  atomicAdd(p + 0, v.x);
  atomicAdd(p + 1, v.y);
  atomicAdd(p + 2, v.z);
  atomicAdd(p + 3, v.w);
}

// ---- fused SAGE linear: out = relu?([agg*inv | x] @ [Wl;Wr] + b) ----------
// One wave32 per 16x16 output tile. N, FIN, FOUT all multiples of 16, so
// EXEC is always all-ones (WMMA requirement) and no edge tiles exist.
//
// V_WMMA_F32_16X16X4_F32 fragment layout (ISA 7.12.2, wave32):
//   A (16x4): lane L -> row (L&15); VGPR0 = K+2*(L>>4), VGPR1 = K+2*(L>>4)+1
//   B (4x16): lane L -> col (L&15); VGPR0 = K+2*(L>>4), VGPR1 = K+2*(L>>4)+1
//   C/D     : VGPR v, lane L -> row v+8*(L>>4), col (L&15)
template <int FIN_T, int FOUT_T, bool RELU>
__global__ __launch_bounds__(32) void sage_gemm_kernel(
    const float* __restrict__ agg, const float* __restrict__ xin,
    const float* __restrict__ deg, const float* __restrict__ Wl,
    const float* __restrict__ Wr, const float* __restrict__ bias,
    float* __restrict__ out) {
  const int lane = threadIdx.x;       // 0..31
  const int half = lane >> 4;         // 0 | 1
  const int l15  = lane & 15;
  const int row  = blockIdx.x * 16 + l15;  // A-fragment row for this lane
  const int col  = blockIdx.y * 16 + l15;  // B/D column for this lane
  const int koff = half * 2;

  const float inv = 1.0f / fmaxf(deg[row], 1.0f);  // mean divisor (deg0 -> 1)
  const float* aRow = agg + (long long)row * FIN_T;
  const float* xRow = xin + (long long)row * FIN_T;
  __builtin_prefetch(aRow, 0, 1);  // global_prefetch_b8
  __builtin_prefetch(xRow, 0, 1);

  v8f acc = {0.f, 0.f, 0.f, 0.f, 0.f, 0.f, 0.f, 0.f};

  // K-pass 1: mean-aggregated neighbors through W_l (scale A by inv at load)
#pragma unroll
  for (int k0 = 0; k0 < FIN_T; k0 += 4) {
    v2f a = *(const v2f*)(aRow + k0 + koff);  // 8B aligned: k0+koff even
    a *= inv;
    v2f b;
    b.x = Wl[(k0 + koff) * FOUT_T + col];
    b.y = Wl[(k0 + koff + 1) * FOUT_T + col];
    acc = __builtin_amdgcn_wmma_f32_16x16x4_f32(
        /*neg_a=*/false, a, /*neg_b=*/false, b,
        /*c_mod=*/(short)0, acc, /*reuse_a=*/false, /*reuse_b=*/false);
  }
  // K-pass 2: root/self features through W_r
#pragma unroll
  for (int k0 = 0; k0 < FIN_T; k0 += 4) {
    v2f a = *(const v2f*)(xRow + k0 + koff);
    v2f b;
    b.x = Wr[(k0 + koff) * FOUT_T + col];
    b.y = Wr[(k0 + koff + 1) * FOUT_T + col];
    acc = __builtin_amdgcn_wmma_f32_16x16x4_f32(
        false, a, false, b, (short)0, acc, false, false);
  }

  const float bb = bias[col];
#pragma unroll
  for (int v = 0; v < 8; ++v) {
    int m = blockIdx.x * 16 + v + half * 8;
    float val = acc[v] + bb;
    if (RELU) val = fmaxf(val, 0.f);
    out[(long long)m * FOUT_T + col] = val;
  }
}

// ---------------------------------------------------------------------------
extern "C" void kernel_launch(void* const* d_in, const int* in_sizes, int n_in,
                              void* d_out, int out_size, void* d_ws,
                              size_t ws_size, hipStream_t stream) {
  constexpr int FIN = 64, HID = 128, FOUT = 32;

  const float* x   = (const float*)d_in[0];
  const int*   ei  = (const int*)d_in[1];  // edge_index flat [2,E]
  const float* Wl1 = (const float*)d_in[2];
  const float* Wr1 = (const float*)d_in[3];
  const float* b1  = (const float*)d_in[4];
  const float* Wl2 = (const float*)d_in[5];
  const float* Wr2 = (const float*)d_in[6];
  const float* b2  = (const float*)d_in[7];
  float* out = (float*)d_out;

  const int n = in_sizes[0] / FIN;  // 100000 (multiple of 16)
  const int E = in_sizes[1] / 2;    // 1600000
  const int* src = ei;
  const int* dst = ei + E;

  // Workspace layout: deg[n] | agg1[n*FIN] | agg2[n*HID] | h[n*HID]
  float* deg  = (float*)d_ws;
  float* agg1 = deg + n;
  float* agg2 = agg1 + (size_t)n * FIN;
  float* h    = agg2 + (size_t)n * HID;

  // Zero deg + agg1 + agg2 (h is fully overwritten by gemm1).
  hipMemsetAsync(d_ws, 0, (size_t)n * (1 + FIN + HID) * sizeof(float), stream);

  sage_deg_kernel<<<(E + 255) / 256, 256, 0, stream>>>(dst, deg, E);

  sage_scatter_kernel<FIN>
      <<<(E * (FIN / 4) + 255) / 256, 256, 0, stream>>>(x, src, dst, agg1, E);

  sage_gemm_kernel<FIN, HID, true>
      <<<dim3(n / 16, HID / 16), 32, 0, stream>>>(agg1, x, deg, Wl1, Wr1, b1, h);

  sage_scatter_kernel<HID>
      <<<(E * (HID / 4) + 255) / 256, 256, 0, stream>>>(h, src, dst, agg2, E);

  sage_gemm_kernel<HID, FOUT, false>
      <<<dim3(n / 16, FOUT / 16), 32, 0, stream>>>(agg2, h, deg, Wl2, Wr2, b2, out);
}